// PointPillarsScatter_81509889344183
// MI455X (gfx1250) — compile-verified
//
#include <hip/hip_runtime.h>
#include <hip/hip_bf16.h>
#include <stdint.h>

// PointPillarsScatter for MI455X (gfx1250).
// Phase 1: zero 256MB BEV map. Per thread: 16 fully-unrolled global_store_b128
//          with immediate offsets (stride 4096B) -> 1 address calc per 256B
//          stored, perfectly coalesced per instruction, RT hint keeps lines in
//          the 192MB L2 for phase 2.
// Phase 2: one wave32 per pillar. Bulk-stage each block's pillar features
//          (8 x 64 f32 = 2KB) and coords (8 x 3 i32 = 96B) into LDS with
//          gfx1250 async global->LDS B128 transfers (ASYNCcnt), then scatter
//          two f32 channels per lane with global_store_b32.

#define NXc 500
#define NYc 500
#define Bc  4
#define Cc  64
#define PPB 8      // pillars per 256-thread block (1 pillar per wave32)
#define ZT  256    // zero-kernel threads per block
#define ZU  16     // float4 stores per thread in zero kernel

__device__ __forceinline__ void async_g2lds_b128(uint32_t lds_byte_off, const void* gptr) {
    asm volatile("global_load_async_to_lds_b128 %0, %1, off"
                 :: "v"(lds_byte_off), "v"(gptr)
                 : "memory");
}

__device__ __forceinline__ void wait_asynccnt0() {
    asm volatile("s_wait_asynccnt 0x0" ::: "memory");
}

// ---------------------------------------------------------------- zero fill
__global__ void bev_zero_kernel(float4* __restrict__ out4, long long n4,
                                float* __restrict__ out, long long n) {
    long long base = (long long)blockIdx.x * (ZT * ZU) + threadIdx.x;
    float4 z; z.x = 0.f; z.y = 0.f; z.z = 0.f; z.w = 0.f;
    if (base + (long long)(ZU - 1) * ZT < n4) {
        // Full tile: 16 stores, immediate offsets 0,4096,...,61440 bytes.
        float4* p = out4 + base;
#pragma unroll
        for (int k = 0; k < ZU; ++k)
            p[k * ZT] = z;                       // global_store_b128 off:k*4096
    } else {
        // Partial tile (hit by the last block for 64M-float output).
#pragma unroll
        for (int k = 0; k < ZU; ++k) {
            long long idx = base + (long long)k * ZT;
            if (idx < n4) out4[idx] = z;
        }
        // Scalar tail if out_size % 4 != 0 (empty here).
        if (base == (long long)blockIdx.x * (ZT * ZU)) {
            for (long long t = n4 * 4 + (long long)threadIdx.x; t < n; t += ZT)
                out[t] = 0.f;
        }
    }
}

// ----------------------------------------------------------------- scatter
__global__ void pillar_scatter_kernel(const float* __restrict__ feat,   // (B*P, C)
                                      const int*   __restrict__ coords, // (B*P, 3)
                                      float* __restrict__ out,          // (B, C, NX, NY)
                                      int numPillars) {
    __shared__ float sfeat[PPB * Cc];   // 512 floats, 2KB
    __shared__ int   scoord[PPB * 3];   // 96 bytes

    const int tid = threadIdx.x;        // 0..255
    const int pb  = blockIdx.x * PPB;   // first pillar of this block

    // ---- async stage: features. PPB*Cc floats = 128 float4 chunks; loop spans 512 (256 thr x2).
    const float* gfeat = feat + (size_t)pb * Cc;
#pragma unroll
    for (int k = 0; k < 2; ++k) {
        int idx = tid + k * 256;                  // float4 chunk index 0..511
        int gp  = pb + (idx >> 4);                // 16 float4 chunks per pillar
        if (gp < numPillars) {
            uint32_t loff = (uint32_t)(uintptr_t)(&sfeat[idx * 4]);
            async_g2lds_b128(loff, gfeat + (size_t)idx * 4);
        }
    }
    // ---- async stage: coords. PPB*3 ints = 96B = 6 x B128.
    if (tid < (PPB * 3) / 4) {
        long long elt = (long long)pb * 3 + tid * 4;
        if (elt + 4 <= (long long)numPillars * 3) {
            uint32_t loff = (uint32_t)(uintptr_t)(&scoord[tid * 4]);
            async_g2lds_b128(loff, coords + elt);
        }
    }

    wait_asynccnt0();
    __syncthreads();

    const int w    = tid >> 5;   // wave id = pillar-in-block
    const int lane = tid & 31;
    const int g    = pb + w;
    if (g >= numPillars) return;

    int bi = scoord[w * 3 + 0];
    int xi = scoord[w * 3 + 1];
    int yi = scoord[w * 3 + 2];

    // Reference semantics: invalid pillars (x==0 && y==0) are dropped.
    if (!((xi > 0) || (yi > 0))) return;

    bi = min(max(bi, 0), Bc - 1);
    xi = min(max(xi, 0), NXc - 1);
    yi = min(max(yi, 0), NYc - 1);

    const int c0 = lane * 2;                              // channels c0, c0+1
    float2 f = *(const float2*)&sfeat[w * Cc + c0];       // ds_load_b64 (8B aligned)

    size_t base = (((size_t)bi * Cc + c0) * NXc + xi) * (size_t)NYc + yi;
    out[base]                         = f.x;              // global_store_b32
    out[base + (size_t)NXc * NYc]     = f.y;              // channel c0+1 plane
}

// ----------------------------------------------------------------- launcher
extern "C" void kernel_launch(void* const* d_in, const int* in_sizes, int n_in,
                              void* d_out, int out_size, void* d_ws, size_t ws_size,
                              hipStream_t stream) {
    const float* feat   = (const float*)d_in[0];   // (B, P, C) f32, flat
    const int*   coords = (const int*)d_in[1];     // (B, P, 3) i32, flat
    float* out = (float*)d_out;                    // (B, C, NX, NY) f32

    // Phase 1: zero the BEV map. 64M floats -> 16M float4 -> 3907 blocks (last one partial).
    long long n  = (long long)out_size;
    long long n4 = n >> 2;
    long long zblocks = (n4 + (long long)(ZT * ZU) - 1) / (ZT * ZU);
    bev_zero_kernel<<<(int)zblocks, ZT, 0, stream>>>((float4*)out, n4, out, n);

    // Phase 2: scatter pillars.
    int numPillars = in_sizes[1] / 3;              // B*P = 48000
    int blocks = (numPillars + PPB - 1) / PPB;     // 6000
    pillar_scatter_kernel<<<blocks, 256, 0, stream>>>(feat, coords, out, numPillars);
}